// GATLayer_25271587569842
// MI455X (gfx1250) — compile-verified
//
#include <hip/hip_runtime.h>
#include <hip/hip_bf16.h>

#define N_NODES 20000
#define N_EDGES 640000
#define D_NODE  128
#define D_EDGE  64
#define D_OUT   128

typedef __attribute__((ext_vector_type(2))) float v2f;
typedef __attribute__((ext_vector_type(8))) float v8f;

// ---------- helpers: order-preserving float<->uint encoding for atomicMax ----------
__device__ __forceinline__ unsigned enc_f32(float f) {
    unsigned u = __float_as_uint(f);
    return (u & 0x80000000u) ? ~u : (u | 0x80000000u);
}
__device__ __forceinline__ float dec_f32(unsigned k) {
    return (k & 0x80000000u) ? __uint_as_float(k & 0x7fffffffu)
                             : __uint_as_float(~k);
}

// ---------- K0: init output + per-node accumulators ----------
__global__ void gat_init_kernel(float* __restrict__ out, float* __restrict__ ssum,
                                unsigned* __restrict__ m_enc) {
    int i = blockIdx.x * blockDim.x + threadIdx.x;
    if (i < N_NODES * D_OUT) out[i] = 0.0f;
    if (i < N_NODES) { ssum[i] = 0.0f; m_enc[i] = 0u; }  // 0 < enc(x) for all finite x
}

// ---------- K1: ae_proj[k] = sum_d We[d][k] * a_e[d]  (We is D_OUT x D_EDGE row-major) ----------
__global__ void gat_aeproj_kernel(const float* __restrict__ We, const float* __restrict__ Wa,
                                  float* __restrict__ aep) {
    int k = threadIdx.x;  // 64 threads
    if (k < D_EDGE) {
        float acc = 0.0f;
        const float* ae = Wa + 2 * D_OUT;
        for (int d = 0; d < D_OUT; ++d) acc += We[d * D_EDGE + k] * ae[d];
        aep[k] = acc;
    }
}

// ---------- K2: z_node = X @ Wn^T via V_WMMA_F32_16X16X4_F32 ----------
// grid = N_NODES/16 blocks, 256 threads (8 waves); wave w handles column tile j0 = 16*w
__global__ __launch_bounds__(256) void gat_znode_wmma_kernel(const float* __restrict__ X,
                                                             const float* __restrict__ Wn,
                                                             float* __restrict__ Z) {
    const int wave = threadIdx.x >> 5;
    const int lane = threadIdx.x & 31;
    const int half = lane >> 4;     // 0: lanes 0-15, 1: lanes 16-31
    const int l16  = lane & 15;
    const int i0 = blockIdx.x * 16; // row tile
    const int j0 = wave * 16;       // column tile

    const float* xrow = X + (i0 + l16) * D_NODE;       // A row m = l16
    const float* wrow = Wn + (j0 + l16) * D_NODE;      // B col n = l16 (B[k][n] = Wn[j0+n][k])

    v8f c = {};
    #pragma unroll
    for (int k = 0; k < D_NODE; k += 4) {
        v2f a, b;
        // A 16x4 f32 layout: VGPR v holds K = k + 2*half + v for row m = l16
        a.x = xrow[k + 2 * half + 0];
        a.y = xrow[k + 2 * half + 1];
        // B 4x16 f32 layout: VGPR v holds row K = k + v + 2*half, col n = l16
        b.x = wrow[k + 2 * half + 0];
        b.y = wrow[k + 2 * half + 1];
        c = __builtin_amdgcn_wmma_f32_16x16x4_f32(false, a, false, b,
                                                  (short)0, c, false, false);
    }
    // C/D 16x16 f32: VGPR r -> row M = r + 8*half, col N = l16
    float* zrow = Z + (size_t)i0 * D_OUT + j0;
    #pragma unroll
    for (int r = 0; r < 8; ++r)
        zrow[(size_t)(r + 8 * half) * D_OUT + l16] = c[r];
}

// ---------- K3: per-node scalars s_src = z_node . a_src, s_dst = z_node . a_dst ----------
__global__ void gat_nodedots_kernel(const float* __restrict__ Z, const float* __restrict__ Wa,
                                    float* __restrict__ s_src, float* __restrict__ s_dst) {
    int node = blockIdx.x * (blockDim.x >> 5) + (threadIdx.x >> 5);
    int lane = threadIdx.x & 31;
    if (node >= N_NODES) return;
    const float* z = Z + (size_t)node * D_OUT;
    float a0 = 0.0f, a1 = 0.0f;
    #pragma unroll
    for (int k = lane; k < D_OUT; k += 32) {
        float zv = z[k];
        a0 += zv * Wa[k];
        a1 += zv * Wa[D_OUT + k];
    }
    #pragma unroll
    for (int off = 16; off > 0; off >>= 1) {
        a0 += __shfl_down(a0, off, 32);
        a1 += __shfl_down(a1, off, 32);
    }
    if (lane == 0) { s_src[node] = a0; s_dst[node] = a1; }
}

// ---------- K4: edge logits + leaky_relu + segment max (one wave per edge) ----------
__global__ void gat_edgelogit_kernel(const float* __restrict__ FE, const float* __restrict__ aep,
                                     const float* __restrict__ s_src, const float* __restrict__ s_dst,
                                     const int* __restrict__ src, const int* __restrict__ dst,
                                     float* __restrict__ elog, unsigned* __restrict__ m_enc) {
    int e = blockIdx.x * (blockDim.x >> 5) + (threadIdx.x >> 5);
    int lane = threadIdx.x & 31;
    if (e >= N_EDGES) return;
    const float* fe = FE + (size_t)e * D_EDGE;
    float acc = fe[lane] * aep[lane] + fe[lane + 32] * aep[lane + 32];
    #pragma unroll
    for (int off = 16; off > 0; off >>= 1) acc += __shfl_down(acc, off, 32);
    if (lane == 0) {
        float logit = acc + s_src[src[e]] + s_dst[dst[e]];
        float lr = (logit > 0.0f) ? logit : 0.2f * logit;  // leaky_relu slope 0.2
        elog[e] = lr;
        atomicMax(&m_enc[dst[e]], enc_f32(lr));
    }
}

// ---------- K5: ex = exp(e - m[dst]); segment sum ----------
__global__ void gat_expsum_kernel(const int* __restrict__ dst, const unsigned* __restrict__ m_enc,
                                  float* __restrict__ elog, float* __restrict__ ssum) {
    int e = blockIdx.x * blockDim.x + threadIdx.x;
    if (e >= N_EDGES) return;
    int d = dst[e];
    float m = dec_f32(m_enc[d]);
    float ex = __expf(elog[e] - m);
    elog[e] = ex;
    atomicAdd(&ssum[d], ex);
}

// ---------- K6: h[dst] += alpha * z_node[src]  (one wave per edge, 4 dims/lane) ----------
__global__ void gat_scatter_kernel(const int* __restrict__ src, const int* __restrict__ dst,
                                   const float* __restrict__ Z, const float* __restrict__ elog,
                                   const float* __restrict__ ssum, float* __restrict__ out) {
    int idx = blockIdx.x * blockDim.x + threadIdx.x;
    int e = idx >> 5;
    int lane = idx & 31;
    if (e >= N_EDGES) return;
    int d = dst[e];
    float s = ssum[d];
    float alpha = elog[e] / ((s == 0.0f) ? 1.0f : s);
    const float4 zv = *(const float4*)(Z + (size_t)src[e] * D_OUT + lane * 4);
    float* o = out + (size_t)d * D_OUT + lane * 4;
    atomicAdd(&o[0], alpha * zv.x);
    atomicAdd(&o[1], alpha * zv.y);
    atomicAdd(&o[2], alpha * zv.z);
    atomicAdd(&o[3], alpha * zv.w);
}

extern "C" void kernel_launch(void* const* d_in, const int* in_sizes, int n_in,
                              void* d_out, int out_size, void* d_ws, size_t ws_size,
                              hipStream_t stream) {
    const float* feats_node = (const float*)d_in[0];
    const float* feats_edge = (const float*)d_in[1];
    const float* Wn = (const float*)d_in[2];
    const float* We = (const float*)d_in[3];
    const float* Wa = (const float*)d_in[4];
    const int*   src = (const int*)d_in[5];
    const int*   dst = (const int*)d_in[6];
    float* out = (float*)d_out;

    // workspace layout (floats)
    float* ws = (float*)d_ws;
    float*    Z      = ws;                           // N_NODES*D_OUT
    float*    s_src  = Z + (size_t)N_NODES * D_OUT;  // N_NODES
    float*    s_dst  = s_src + N_NODES;              // N_NODES
    float*    ssum   = s_dst + N_NODES;              // N_NODES
    unsigned* m_enc  = (unsigned*)(ssum + N_NODES);  // N_NODES
    float*    aep    = (float*)(m_enc + N_NODES);    // D_EDGE
    float*    elog   = aep + D_EDGE;                 // N_EDGES

    // K0: init (covers out zeroing + per-node accumulators)
    {
        int n = N_NODES * D_OUT;
        gat_init_kernel<<<(n + 255) / 256, 256, 0, stream>>>(out, ssum, m_enc);
    }
    // K1: fold We into a_e
    gat_aeproj_kernel<<<1, 64, 0, stream>>>(We, Wa, aep);
    // K2: z_node GEMM via WMMA f32 16x16x4
    gat_znode_wmma_kernel<<<N_NODES / 16, 256, 0, stream>>>(feats_node, Wn, Z);
    // K3: per-node attention scalars (8 waves/block)
    gat_nodedots_kernel<<<(N_NODES + 7) / 8, 256, 0, stream>>>(Z, Wa, s_src, s_dst);
    // K4: edge logits + segment max (8 edges/block)
    gat_edgelogit_kernel<<<(N_EDGES + 7) / 8, 256, 0, stream>>>(feats_edge, aep, s_src, s_dst,
                                                                src, dst, elog, m_enc);
    // K5: exp + segment sum
    gat_expsum_kernel<<<(N_EDGES + 255) / 256, 256, 0, stream>>>(dst, m_enc, elog, ssum);
    // K6: weighted scatter into output (32 lanes per edge)
    {
        long long total = (long long)N_EDGES * 32;
        gat_scatter_kernel<<<(int)((total + 255) / 256), 256, 0, stream>>>(src, dst, Z, elog,
                                                                           ssum, out);
    }
}